// AIGEncoder_85925115724498
// MI455X (gfx1250) — compile-verified
//
#include <hip/hip_runtime.h>

typedef float v2f __attribute__((ext_vector_type(2)));
typedef float v8f __attribute__((ext_vector_type(8)));

#define Hdim 128
#define XS   130   // activation tile row stride (even => 8B-aligned A pairs)
#define WP2  144   // weight pair-row stride in float2 units (2*WP2 % 64 == 32)

// ---------------------------------------------------------------------------
// f32 WMMA wrapper: D(16x16) = A(16x4) * B(4x16) + C
// ---------------------------------------------------------------------------
__device__ __forceinline__ v8f wmma_f32_k4(v2f a, v2f b, v8f c) {
    return __builtin_amdgcn_wmma_f32_16x16x4_f32(
        /*neg_a=*/false, a, /*neg_b=*/false, b,
        /*c_mod=*/(short)0, c, /*reuse_a=*/false, /*reuse_b=*/false);
}

// Cooperative: stage 128x128 weights as K-pair-interleaved float2 in LDS.
// lds_wp[kp*WP2 + n] = { W[2kp][n], W[2kp+1][n] }
__device__ __forceinline__ void load_w_pairs(const float* __restrict__ w,
                                             v2f* lds_wp, int tid) {
    for (int i4 = tid; i4 < 64 * 32; i4 += 256) {
        int kp = i4 >> 5, n0 = (i4 & 31) << 2;
        float4 lo = *(const float4*)(w + (size_t)(2 * kp) * Hdim + n0);
        float4 hi = *(const float4*)(w + (size_t)(2 * kp + 1) * Hdim + n0);
        v2f* p = lds_wp + kp * WP2 + n0;
        p[0] = v2f{lo.x, hi.x};
        p[1] = v2f{lo.y, hi.y};
        p[2] = v2f{lo.z, hi.z};
        p[3] = v2f{lo.w, hi.w};
    }
}

// Per-wave GEMM: 16 rows (wm0..wm0+15) x 128 cols, K=128.
// A from lds_a (row stride XS, b64 loads), B from paired lds_wp (b64 loads).
__device__ __forceinline__ void wave_gemm(const float* lds_a, const v2f* lds_wp,
                                          const float* __restrict__ bias,
                                          int wm0, int lane, v8f acc[8]) {
    const int l15  = lane & 15;
    const int koff = lane >> 4;   // 0 for lanes 0-15, 1 (pair) for 16-31
#pragma unroll
    for (int nt = 0; nt < 8; ++nt) {
        float b = bias[nt * 16 + l15];
        acc[nt] = v8f{b, b, b, b, b, b, b, b};
    }
    const float* arow = lds_a + (wm0 + l15) * XS;
    for (int k = 0; k < Hdim; k += 4) {
        v2f a = *(const v2f*)__builtin_assume_aligned(arow + k + (koff << 1), 8);
        const v2f* wrow = lds_wp + ((k >> 1) + koff) * WP2;
#pragma unroll
        for (int nt = 0; nt < 8; ++nt) {
            v2f bf = wrow[nt * 16 + l15];
            acc[nt] = wmma_f32_k4(a, bf, acc[nt]);
        }
    }
}

// Scatter a wave's 16x128 accumulator tile (C layout) into LDS row-major.
__device__ __forceinline__ void store_tile(float* lds_x, int wm0, int lane,
                                           const v8f acc[8], bool do_relu) {
    const int l15   = lane & 15;
    const int mhalf = (lane >> 4) << 3;   // +8 rows for lanes 16-31
#pragma unroll
    for (int nt = 0; nt < 8; ++nt) {
#pragma unroll
        for (int r = 0; r < 8; ++r) {
            float v = acc[nt][r];
            if (do_relu) v = fmaxf(v, 0.f);
            lds_x[(wm0 + r + mhalf) * XS + nt * 16 + l15] = v;
        }
    }
}

// ---------------------------------------------------------------------------
// Edge message passing, din == 5 (layer 0): one thread per edge.
// ---------------------------------------------------------------------------
__global__ __launch_bounds__(256) void msg_din5(
    const float* __restrict__ x, const int* __restrict__ ei,
    const float* __restrict__ ea, const float* __restrict__ ew,
    const float* __restrict__ eb, float* __restrict__ aggr, int nE) {
    int e = blockIdx.x * 256 + threadIdx.x;
    if (e >= nE) return;
    int s = ei[e];
    int d = ei[nE + e];
    float a = ea[e];
#pragma unroll
    for (int k = 0; k < 5; ++k) {
        float m = fmaxf(x[(size_t)s * 5 + k] + (a * ew[k] + eb[k]), 0.f);
        atomicAdd(&aggr[(size_t)d * 5 + k], m);
    }
}

// ---------------------------------------------------------------------------
// Edge message passing, din == 128: 32 lanes per edge (4 cols each).
// ---------------------------------------------------------------------------
__global__ __launch_bounds__(256) void msg_h(
    const float* __restrict__ x, const int* __restrict__ ei,
    const float* __restrict__ ea, const float* __restrict__ ew,
    const float* __restrict__ eb, float* __restrict__ aggr, int nE) {
    int t = blockIdx.x * 256 + threadIdx.x;
    if (t >= nE * 32) return;
    int e  = t >> 5;
    int c4 = (t & 31) << 2;
    int s = ei[e];
    int d = ei[nE + e];
    float a = ea[e];
    const float4 xv = *(const float4*)(x + (size_t)s * Hdim + c4);
    const float4 wv = *(const float4*)(ew + c4);
    const float4 bv = *(const float4*)(eb + c4);
    float m0 = fmaxf(xv.x + a * wv.x + bv.x, 0.f);
    float m1 = fmaxf(xv.y + a * wv.y + bv.y, 0.f);
    float m2 = fmaxf(xv.z + a * wv.z + bv.z, 0.f);
    float m3 = fmaxf(xv.w + a * wv.w + bv.w, 0.f);
    float* ap = aggr + (size_t)d * Hdim + c4;
    atomicAdd(ap + 0, m0);
    atomicAdd(ap + 1, m1);
    atomicAdd(ap + 2, m2);
    atomicAdd(ap + 3, m3);
}

// ---------------------------------------------------------------------------
// Fused MLP + LayerNorm + ReLU. Block = 256 threads (8 waves), 128 rows.
//   h = x + aggr; T = relu(h@W1+b1); Y = T@W2+b2; out = relu(LN(Y)*g+bt)
// Safe with out == x (rows fully staged into LDS before any global write).
// ---------------------------------------------------------------------------
template <int DIN>
__global__ __launch_bounds__(256) void mlp_ln_kernel(
    const float* __restrict__ x, const float* __restrict__ aggr,
    const float* __restrict__ w1, const float* __restrict__ b1,
    const float* __restrict__ w2, const float* __restrict__ b2,
    const float* __restrict__ gam, const float* __restrict__ bet,
    float* __restrict__ out, int n) {
    extern __shared__ float lds[];
    v2f*   lds_wp = (v2f*)lds;                  // 64 * WP2 float2
    float* lds_x  = lds + 2 * 64 * WP2;         // 128 x XS
    float* lds_mu = lds_x + 128 * XS;           // 128
    float* lds_rs = lds_mu + 128;               // 128
    float* lds_h  = lds_rs + 128;               // 128 x 8 (DIN==5 path)

    const int tid  = threadIdx.x;
    const int lane = tid & 31;
    const int wave = tid >> 5;
    const int wm0  = wave * 16;
    const int row0 = blockIdx.x * 128;

    if constexpr (DIN == Hdim) {
        load_w_pairs(w1, lds_wp, tid);
        for (int i4 = tid; i4 < 128 * Hdim / 4; i4 += 256) {
            int base = i4 * 4, r = base >> 7, c = base & 127;
            int row = row0 + r;
            float4 v = {0.f, 0.f, 0.f, 0.f};
            if (row < n) {
                float4 xv = *(const float4*)(x + (size_t)row * Hdim + c);
                float4 av = *(const float4*)(aggr + (size_t)row * Hdim + c);
                v.x = xv.x + av.x; v.y = xv.y + av.y;
                v.z = xv.z + av.z; v.w = xv.w + av.w;
            }
            float* p = lds_x + r * XS + c;
            p[0] = v.x; p[1] = v.y; p[2] = v.z; p[3] = v.w;
        }
        __syncthreads();
        v8f acc[8];
        wave_gemm(lds_x, lds_wp, b1, wm0, lane, acc);
        store_tile(lds_x, wm0, lane, acc, /*relu=*/true);  // own rows only
    } else {
        // DIN == 5: stage W1 (5x128, flat) + h tile (128x5), scalar GEMM1.
        float* lds_wf = (float*)lds_wp;
        for (int i = tid; i < DIN * Hdim; i += 256) lds_wf[i] = w1[i];
        for (int i = tid; i < 128 * DIN; i += 256) {
            int r = i / DIN, k = i - r * DIN;
            int row = row0 + r;
            lds_h[i] = (row < n) ? (x[(size_t)row * DIN + k] + aggr[(size_t)row * DIN + k]) : 0.f;
        }
        __syncthreads();
        for (int i = tid; i < 128 * Hdim; i += 256) {
            int m = i >> 7, c = i & 127;
            float v = b1[c];
#pragma unroll
            for (int k = 0; k < DIN; ++k) v += lds_h[m * DIN + k] * lds_wf[k * Hdim + c];
            lds_x[m * XS + c] = fmaxf(v, 0.f);
        }
    }
    __syncthreads();  // W1 reads done -> reuse lds_wp for W2

    load_w_pairs(w2, lds_wp, tid);
    __syncthreads();

    v8f acc2[8];
    wave_gemm(lds_x, lds_wp, b2, wm0, lane, acc2);
    store_tile(lds_x, wm0, lane, acc2, /*relu=*/false);  // own rows only
    __syncthreads();

    // LayerNorm row stats: one thread per row (even stride => no conflicts).
    if (tid < 128) {
        const float* rp = (const float*)__builtin_assume_aligned(lds_x + tid * XS, 8);
        float s = 0.f;
        for (int c = 0; c < Hdim; ++c) s += rp[c];
        float mu = s * (1.f / Hdim);
        float vs = 0.f;
        for (int c = 0; c < Hdim; ++c) { float dd = rp[c] - mu; vs += dd * dd; }
        lds_mu[tid] = mu;
        lds_rs[tid] = rsqrtf(vs * (1.f / Hdim) + 1e-5f);
    }
    __syncthreads();

    for (int i4 = tid; i4 < 128 * Hdim / 4; i4 += 256) {
        int base = i4 * 4, r = base >> 7, c = base & 127;
        int row = row0 + r;
        if (row >= n) continue;
        float mu = lds_mu[r], rs = lds_rs[r];
        float4 gv = *(const float4*)(gam + c);
        float4 bv = *(const float4*)(bet + c);
        const float* p = lds_x + r * XS + c;
        float4 o;
        o.x = fmaxf((p[0] - mu) * rs * gv.x + bv.x, 0.f);
        o.y = fmaxf((p[1] - mu) * rs * gv.y + bv.y, 0.f);
        o.z = fmaxf((p[2] - mu) * rs * gv.z + bv.z, 0.f);
        o.w = fmaxf((p[3] - mu) * rs * gv.w + bv.w, 0.f);
        *(float4*)(out + (size_t)row * Hdim + c) = o;
    }
}

// ---------------------------------------------------------------------------
// Segment-sum pooling over sorted `batch`: 128 threads = 128 cols, each block
// scans a contiguous node chunk, flushing with atomics only at group edges.
// ---------------------------------------------------------------------------
__global__ __launch_bounds__(128) void pool_accum(
    const float* __restrict__ x, const int* __restrict__ batch,
    float* __restrict__ pool, float* __restrict__ cnt, int n, int chunk) {
    int c = threadIdx.x;
    int start = blockIdx.x * chunk;
    int end = start + chunk;
    if (end > n) end = n;
    if (start >= end) return;
    int curg = batch[start];
    float acc = 0.f, cc = 0.f;
    for (int i = start; i < end; ++i) {
        int g = batch[i];
        if (g != curg) {
            atomicAdd(&pool[curg * Hdim + c], acc);
            if (c == 0) atomicAdd(&cnt[curg], cc);
            acc = 0.f; cc = 0.f; curg = g;
        }
        acc += x[(size_t)i * Hdim + c];
        cc += 1.f;
    }
    atomicAdd(&pool[curg * Hdim + c], acc);
    if (c == 0) atomicAdd(&cnt[curg], cc);
}

__global__ __launch_bounds__(256) void pool_final(
    const float* __restrict__ pool, const float* __restrict__ cnt,
    float* __restrict__ out, int nG) {
    int t = blockIdx.x * 256 + threadIdx.x;
    if (t >= nG * 2 * Hdim) return;
    int g = t >> 8, j = t & 255;
    float s = pool[g * Hdim + (j & 127)];
    out[(size_t)g * 256 + j] = (j < 128) ? s / fmaxf(cnt[g], 1.f) : s;
}

// ---------------------------------------------------------------------------
extern "C" void kernel_launch(void* const* d_in, const int* in_sizes, int n_in,
                              void* d_out, int out_size, void* d_ws, size_t ws_size,
                              hipStream_t stream) {
    const int N = 200000, E = 400000, G = 64, IN = 5;
    const float* x0    = (const float*)d_in[0];
    const int*   ei    = (const int*)d_in[1];
    const float* ea    = (const float*)d_in[2];
    const int*   batch = (const int*)d_in[3];
    auto L = [&](int l, int j) { return (const float*)d_in[4 + l * 8 + j]; };
    // j: 0=ew 1=eb 2=w1 3=b1 4=w2 5=b2 6=g 7=bt

    float* ws   = (float*)d_ws;
    float* aggr = ws;                           // N*128
    float* bufB = aggr + (size_t)N * Hdim;      // N*128 (feature buffer)
    float* pool = bufB + (size_t)N * Hdim;      // G*128
    float* cnt  = pool + G * Hdim;              // G

    const size_t ldsBytes  = (size_t)(2 * 64 * WP2 + 128 * XS + 2 * 128 + 128 * 8) * 4;
    const int    mlpBlocks = (N + 127) / 128;
    const int    msgHBlk   = (E * 32 + 255) / 256;

    // ---- layer 0 (din = 5) ----
    hipMemsetAsync(aggr, 0, (size_t)N * IN * sizeof(float), stream);
    msg_din5<<<(E + 255) / 256, 256, 0, stream>>>(x0, ei, ea, L(0, 0), L(0, 1), aggr, E);
    mlp_ln_kernel<5><<<mlpBlocks, 256, ldsBytes, stream>>>(
        x0, aggr, L(0, 2), L(0, 3), L(0, 4), L(0, 5), L(0, 6), L(0, 7), bufB, N);

    // ---- layer 1 (din = 128), MLP in-place on bufB ----
    hipMemsetAsync(aggr, 0, (size_t)N * Hdim * sizeof(float), stream);
    msg_h<<<msgHBlk, 256, 0, stream>>>(bufB, ei, ea, L(1, 0), L(1, 1), aggr, E);
    mlp_ln_kernel<Hdim><<<mlpBlocks, 256, ldsBytes, stream>>>(
        bufB, aggr, L(1, 2), L(1, 3), L(1, 4), L(1, 5), L(1, 6), L(1, 7), bufB, N);

    // ---- layer 2 (din = 128), MLP in-place on bufB ----
    hipMemsetAsync(aggr, 0, (size_t)N * Hdim * sizeof(float), stream);
    msg_h<<<msgHBlk, 256, 0, stream>>>(bufB, ei, ea, L(2, 0), L(2, 1), aggr, E);
    mlp_ln_kernel<Hdim><<<mlpBlocks, 256, ldsBytes, stream>>>(
        bufB, aggr, L(2, 2), L(2, 3), L(2, 4), L(2, 5), L(2, 6), L(2, 7), bufB, N);

    // ---- pooling ----
    hipMemsetAsync(pool, 0, (size_t)(G * Hdim + G) * sizeof(float), stream);
    const int chunk = 256;
    pool_accum<<<(N + chunk - 1) / chunk, 128, 0, stream>>>(bufB, batch, pool, cnt, N, chunk);
    pool_final<<<(G * 2 * Hdim + 255) / 256, 256, 0, stream>>>(pool, cnt, (float*)d_out, G);
}